// Encoder_24146306138770
// MI455X (gfx1250) — compile-verified
//
#include <hip/hip_runtime.h>
#include <hip/hip_bf16.h>

// ---------------------------------------------------------------------------
// 3-layer bidirectional GRU for MI455X (gfx1250), wave32 + V_WMMA_F32_16X16X4_F32
//
//   * Input projections (no recurrence) = big WMMA GEMMs over all 32768 rows.
//   * Recurrence = persistent single-workgroup scan kernel; W_hh^T + h + gh
//     live in LDS (307KB of the WGP's 320KB).
//   * 480-thread blocks = 15 wave32 = exactly one wave per 16-col tile of the
//     225-wide gate matrix -> guard-free, fixed-register accumulators.
// ---------------------------------------------------------------------------

typedef __attribute__((ext_vector_type(2))) float v2f;
typedef __attribute__((ext_vector_type(8))) float v8f;

#define LSEQ   512
#define BATCH  64
#define MROWS  (LSEQ * BATCH)   // 32768
#define HID    75
#define NG     225              // 3*H
#define NGP    240              // 15 WMMA column tiles
#define KHP    76               // H padded to multiple of 4
#define XLD    152              // 2H padded to multiple of 4
#define CHUNK_K 76
#define TPB    480              // 15 waves: one wave per column tile

// ---------------------------------------------------------------------------
// Input projection:  G[d][row][n] = sum_k X[row][k] * W[d][n][k] + bias[d][n]
// grid = (MROWS/64, 2 dirs), block = 480
// LDS: Wt chunk 76x240 + Xs chunk 64x76 = 92,416 B
// ---------------------------------------------------------------------------
__global__ __launch_bounds__(TPB)
void gru_inproj_wmma(const float* __restrict__ X, int lda, int Kpad, int Kw,
                     const float* __restrict__ W, long wStride,
                     const float* __restrict__ bias, long bStride,
                     float* __restrict__ G, long gStride)
{
    extern __shared__ float lds[];
    float* Wt = lds;                        // CHUNK_K x NGP
    float* Xs = Wt + CHUNK_K * NGP;         // BATCH x CHUNK_K

    const int d    = blockIdx.y;
    const float* Wd = W + (long)d * wStride;
    const float* bd = bias + (long)d * bStride;
    float*       Gd = G + (long)d * gStride;

    const int tid  = threadIdx.x;
    const int ct   = tid >> 5;              // wave id == column tile (0..14)
    const int lane = tid & 31;
    const int half = lane >> 4;             // 0: K0/K1 pair, 1: K2/K3 pair
    const int m    = lane & 15;
    const int row0 = blockIdx.x * 64;

    v8f acc[4];
#pragma unroll
    for (int rt = 0; rt < 4; ++rt)
#pragma unroll
        for (int e = 0; e < 8; ++e) acc[rt][e] = 0.0f;

    const int nChunks = (Kpad + CHUNK_K - 1) / CHUNK_K;
    for (int c = 0; c < nChunks; ++c) {
        const int k0c = c * CHUNK_K;
        // stage W^T chunk (zero-padded in k and n)
        for (int i = tid; i < CHUNK_K * NGP; i += TPB) {
            const int kk = i / NGP, n = i % NGP;
            const int k = k0c + kk;
            Wt[i] = (k < Kw && n < NG) ? Wd[(long)n * Kw + k] : 0.0f;
        }
        // stage X chunk (zero-padded in k)
        for (int i = tid; i < BATCH * CHUNK_K; i += TPB) {
            const int b = i / CHUNK_K, kk = i % CHUNK_K;
            const int k = k0c + kk;
            Xs[i] = (k < Kw) ? X[(long)(row0 + b) * lda + k] : 0.0f;
        }
        __syncthreads();

        const float* Bp = Wt + ct * 16 + m;
#pragma unroll
        for (int kk = 0; kk < CHUNK_K; kk += 4) {
            const int kb = kk + half * 2;
            v2f b;
            b.x = Bp[kb * NGP];
            b.y = Bp[(kb + 1) * NGP];
#pragma unroll
            for (int rt = 0; rt < 4; ++rt) {
                const float* ar = Xs + (rt * 16 + m) * CHUNK_K;
                v2f a; a.x = ar[kb]; a.y = ar[kb + 1];
                acc[rt] = __builtin_amdgcn_wmma_f32_16x16x4_f32(
                    false, a, false, b, (short)0, acc[rt], false, false);
            }
        }
        __syncthreads();
    }

    // store + bias (C/D layout: VGPR v holds rows v and v+8 across lane halves)
    const int n = ct * 16 + m;
    if (n < NG) {
        const float bv = bd[n];
#pragma unroll
        for (int rt = 0; rt < 4; ++rt) {
#pragma unroll
            for (int v = 0; v < 8; ++v) {
                const int row = row0 + rt * 16 + half * 8 + v;
                Gd[(long)row * NG + n] = acc[rt][v] + bv;
            }
        }
    }
}

// ---------------------------------------------------------------------------
// Recurrent scan: one persistent workgroup (480 threads, 15 wave32) runs all
// 512 timesteps for one layer, both directions.
// LDS: Wt[2][76][240] + h[2][64][76] + gh[2][64][240]  = 307,712 B
// ---------------------------------------------------------------------------
__global__ __launch_bounds__(TPB)
void gru_scan_wmma(const float* __restrict__ G,      // [2][MROWS][225] gi(+b_ih)
                   const float* __restrict__ Whh, long whStride,  // [d](225x75)
                   const float* __restrict__ bhh, long bhStride,  // [d](225)
                   const float* __restrict__ h0,     // [2][64][75]
                   float* __restrict__ out,          // + layer offset, stride 450
                   float* __restrict__ Xout,         // stride 152 (or null)
                   int Lsteps)
{
    extern __shared__ float lds[];
    float* Wt    = lds;                              // 2*76*240
    float* hBuf  = Wt + 2 * KHP * NGP;               // 2*64*76
    float* ghBuf = hBuf + 2 * BATCH * KHP;           // 2*64*240

    const int tid  = threadIdx.x;
    const int ct   = tid >> 5;                       // wave id == column tile
    const int lane = tid & 31;
    const int half = lane >> 4;
    const int m    = lane & 15;

    // stage transposed, zero-padded W_hh
    for (int i = tid; i < 2 * KHP * NGP; i += TPB) {
        const int d = i / (KHP * NGP), r = i % (KHP * NGP);
        const int kk = r / NGP, n = r % NGP;
        Wt[i] = (kk < HID && n < NG)
                    ? Whh[(long)d * whStride + (long)n * HID + kk] : 0.0f;
    }
    // stage initial hidden state (k padded with zero)
    for (int i = tid; i < 2 * BATCH * KHP; i += TPB) {
        const int d = i / (BATCH * KHP), r = i % (BATCH * KHP);
        const int b = r / KHP, k = r % KHP;
        hBuf[i] = (k < HID) ? h0[((long)d * BATCH + b) * HID + k] : 0.0f;
    }
    __syncthreads();

    for (int t = 0; t < Lsteps; ++t) {
        // prefetch next step's precomputed gates (global_prefetch_b8)
        if (t + 1 < Lsteps) {
            const long nbase = (long)(t + 1) * BATCH * NG;
            for (int p = tid; p < 900; p += TPB) {       // 2*450 cachelines
                const int dd = p / 450, off = p % 450;
                __builtin_prefetch(G + (long)dd * MROWS * NG + nbase + off * 32, 0, 0);
            }
        }

        // gh = h @ W_hh^T : per wave, column tile ct for both dirs x 4 row tiles
#pragma unroll
        for (int d = 0; d < 2; ++d) {
            const float* hA  = hBuf + d * BATCH * KHP;
            const float* Bp  = Wt + d * KHP * NGP + ct * 16 + m;
            v8f acc[4];
#pragma unroll
            for (int rt = 0; rt < 4; ++rt)
#pragma unroll
                for (int e = 0; e < 8; ++e) acc[rt][e] = 0.0f;
#pragma unroll
            for (int kk = 0; kk < KHP; kk += 4) {
                const int kb = kk + half * 2;
                v2f b;
                b.x = Bp[kb * NGP];
                b.y = Bp[(kb + 1) * NGP];
#pragma unroll
                for (int rt = 0; rt < 4; ++rt) {
                    const float* hr = hA + (rt * 16 + m) * KHP;
                    v2f a; a.x = hr[kb]; a.y = hr[kb + 1];
                    acc[rt] = __builtin_amdgcn_wmma_f32_16x16x4_f32(
                        false, a, false, b, (short)0, acc[rt], false, false);
                }
            }
#pragma unroll
            for (int rt = 0; rt < 4; ++rt) {
                float* ghT = ghBuf + d * BATCH * NGP
                           + (rt * 16 + half * 8) * NGP + ct * 16 + m;
#pragma unroll
                for (int v = 0; v < 8; ++v) ghT[v * NGP] = acc[rt][v];
            }
        }
        __syncthreads();

        // gates + state update: 2*64*75 = 9600 elements
        for (int i = tid; i < 2 * BATCH * HID; i += TPB) {
            const int d = i / (BATCH * HID), r = i % (BATCH * HID);
            const int b = r / HID, j = r % HID;
            const long grow = ((long)d * MROWS + (long)t * BATCH + b) * NG;
            const float gir = G[grow + j];
            const float giz = G[grow + HID + j];
            const float gin = G[grow + 2 * HID + j];
            const float* bh = bhh + (long)d * bhStride;
            const float* gh = ghBuf + (d * BATCH + b) * NGP;
            const float hr = gh[j]           + bh[j];
            const float hz = gh[HID + j]     + bh[HID + j];
            const float hn = gh[2 * HID + j] + bh[2 * HID + j];
            const float rg = 1.0f / (1.0f + __expf(-(gir + hr)));
            const float zg = 1.0f / (1.0f + __expf(-(giz + hz)));
            const float ng = tanhf(gin + rg * hn);
            const float hprev = hBuf[(d * BATCH + b) * KHP + j];
            const float hnew  = (1.0f - zg) * ng + zg * hprev;
            hBuf[(d * BATCH + b) * KHP + j] = hnew;
            const long orow = (long)t * BATCH + b;
            out[orow * 450 + d * HID + j] = hnew;
            if (Xout) Xout[orow * XLD + d * HID + j] = hnew;
        }
        __syncthreads();
    }
}

// ---------------------------------------------------------------------------
extern "C" void kernel_launch(void* const* d_in, const int* in_sizes, int n_in,
                              void* d_out, int out_size, void* d_ws, size_t ws_size,
                              hipStream_t stream)
{
    (void)in_sizes; (void)n_in; (void)out_size; (void)ws_size;
    const float* input  = (const float*)d_in[0];   // (512,64,300)
    const float* h0     = (const float*)d_in[1];   // (6,64,75)
    const float* w_ih0  = (const float*)d_in[2];   // (2,225,300)
    const float* w_hh0  = (const float*)d_in[3];   // (2,225,75)
    const float* b_ih0  = (const float*)d_in[4];   // (2,225)
    const float* b_hh0  = (const float*)d_in[5];   // (2,225)
    const float* w_ih12 = (const float*)d_in[6];   // (2,2,225,150)
    const float* w_hh12 = (const float*)d_in[7];   // (2,2,225,75)
    const float* b_ih12 = (const float*)d_in[8];   // (2,2,225)
    const float* b_hh12 = (const float*)d_in[9];   // (2,2,225)
    float* out = (float*)d_out;                    // (512,64,450)

    float* Gbuf = (float*)d_ws;                          // 2*32768*225 f32
    float* Xbuf = Gbuf + (size_t)2 * MROWS * NG;         // 32768*152  f32
    hipMemsetAsync(Xbuf, 0, (size_t)MROWS * XLD * sizeof(float), stream);

    const dim3 gGrid(MROWS / 64, 2), gBlk(TPB);
    const size_t gLds = (size_t)(CHUNK_K * NGP + BATCH * CHUNK_K) * sizeof(float);
    const size_t sLds = (size_t)(2 * KHP * NGP + 2 * BATCH * KHP
                                 + 2 * BATCH * NGP) * sizeof(float);
    const long gStride = (long)MROWS * NG;

    // ---- layer 0 ----
    gru_inproj_wmma<<<gGrid, gBlk, gLds, stream>>>(
        input, 300, 300, 300, w_ih0, 225L * 300, b_ih0, 225, Gbuf, gStride);
    gru_scan_wmma<<<1, TPB, sLds, stream>>>(
        Gbuf, w_hh0, 225L * 75, b_hh0, 225, h0, out + 0, Xbuf, LSEQ);

    // ---- layer 1 ----
    gru_inproj_wmma<<<gGrid, gBlk, gLds, stream>>>(
        Xbuf, XLD, XLD, 150, w_ih12, 225L * 150, b_ih12, 225, Gbuf, gStride);
    gru_scan_wmma<<<1, TPB, sLds, stream>>>(
        Gbuf, w_hh12, 225L * 75, b_hh12, 225,
        h0 + (size_t)2 * BATCH * HID, out + 150, Xbuf, LSEQ);

    // ---- layer 2 ----
    gru_inproj_wmma<<<gGrid, gBlk, gLds, stream>>>(
        Xbuf, XLD, XLD, 150, w_ih12 + (size_t)2 * 225 * 150, 225L * 150,
        b_ih12 + 2 * 225, 225, Gbuf, gStride);
    gru_scan_wmma<<<1, TPB, sLds, stream>>>(
        Gbuf, w_hh12 + (size_t)2 * 225 * 75, 225L * 75,
        b_hh12 + 2 * 225, 225,
        h0 + (size_t)4 * BATCH * HID, out + 300, nullptr, LSEQ);
}